// Sense2VecRRN_35150012351031
// MI455X (gfx1250) — compile-verified
//
#include <hip/hip_runtime.h>
#include <hip/hip_bf16.h>
#include <math.h>

// ---- sizes ----
#define Bq 32
#define Tq 20
#define Vq 32000
#define Hq 100
#define Gq 300          // 3*H
#define BT (Bq*Tq)      // 640
#define KDIM 200        // 2*H  (compile-time K for both WMMA GEMMs)

typedef __attribute__((ext_vector_type(2))) float v2f;
typedef __attribute__((ext_vector_type(8))) float v8f;

// -------------------------------------------------------------------------
// 1) Embedding gather: gx[bt][g] = W_ih[g][x[bt]]  (one-hot matmul == gather)
// -------------------------------------------------------------------------
__global__ void embed_kernel(const int* __restrict__ x,
                             const float* __restrict__ Wf,
                             const float* __restrict__ Wb,
                             float* __restrict__ gxf,
                             float* __restrict__ gxb) {
    int i = blockIdx.x * blockDim.x + threadIdx.x;   // over BT*Gq
    if (i >= BT * Gq) return;
    int bt = i / Gq;
    int g  = i % Gq;
    int tok = x[bt];
    gxf[i] = Wf[(size_t)g * Vq + tok];
    gxb[i] = Wb[(size_t)g * Vq + tok];
}

// -------------------------------------------------------------------------
// 2) GRU scan, one workgroup per direction (blockIdx.x = dir; 1=reverse).
//    h, gh, h_new live in LDS (64,000 B). PyTorch gate order r,z,n; no bias.
// -------------------------------------------------------------------------
__global__ void __launch_bounds__(512)
gru_kernel(const float* __restrict__ gx_f, const float* __restrict__ gx_b,
           const float* __restrict__ Whh_f, const float* __restrict__ Whh_b,
           const float* __restrict__ h0_base,   // hidden + layer*2*B*H
           float* __restrict__ y,               // [BT][2H], col offset dir*H
           float* __restrict__ hT_base) {       // d_out tail + layer*2*B*H
    const int dir = blockIdx.x;
    const float* gx = dir ? gx_b : gx_f;
    const float* W  = dir ? Whh_b : Whh_f;      // [Gq][Hq] row-major
    const float* h0 = h0_base + dir * Bq * Hq;
    float* hT       = hT_base + dir * Bq * Hq;
    const int ycol  = dir * Hq;
    const int tid   = threadIdx.x;
    const int nthr  = blockDim.x;

    __shared__ float h[Bq * Hq];      // 3200
    __shared__ float gh[Bq * Gq];     // 9600
    __shared__ float hnew[Bq * Hq];   // 3200

    for (int p = tid; p < Bq * Hq; p += nthr) h[p] = h0[p];
    __syncthreads();

    for (int s = 0; s < Tq; ++s) {
        const int t = dir ? (Tq - 1 - s) : s;
        // gh[b][g] = sum_k h[b][k] * W[g][k]   (rows are 16B-aligned: 100*4)
        for (int p = tid; p < Bq * Gq; p += nthr) {
            const int b = p / Gq, g = p % Gq;
            const float4* h4 = reinterpret_cast<const float4*>(&h[b * Hq]);
            const float4* w4 = reinterpret_cast<const float4*>(&W[g * Hq]);
            float acc = 0.0f;
#pragma unroll 5
            for (int k = 0; k < Hq / 4; ++k) {
                const float4 hv = h4[k];
                const float4 wv = w4[k];
                acc = fmaf(hv.x, wv.x, acc);
                acc = fmaf(hv.y, wv.y, acc);
                acc = fmaf(hv.z, wv.z, acc);
                acc = fmaf(hv.w, wv.w, acc);
            }
            gh[p] = acc;
        }
        __syncthreads();
        // gate combine + state update
        for (int p = tid; p < Bq * Hq; p += nthr) {
            const int b = p / Hq, j = p % Hq;
            const float* g0 = &gx[(size_t)(b * Tq + t) * Gq];
            const float xr = g0[j], xz = g0[Hq + j], xn = g0[2 * Hq + j];
            const float hr = gh[b * Gq + j];
            const float hz = gh[b * Gq + Hq + j];
            const float hn = gh[b * Gq + 2 * Hq + j];
            const float r = 1.0f / (1.0f + __expf(-(xr + hr)));
            const float z = 1.0f / (1.0f + __expf(-(xz + hz)));
            const float n = tanhf(xn + r * hn);
            const float hv = (1.0f - z) * n + z * h[p];
            hnew[p] = hv;
            y[(size_t)(b * Tq + t) * (2 * Hq) + ycol + j] = hv;
        }
        __syncthreads();
        for (int p = tid; p < Bq * Hq; p += nthr) h[p] = hnew[p];
        __syncthreads();
    }
    for (int p = tid; p < Bq * Hq; p += nthr) hT[p] = h[p];
}

// -------------------------------------------------------------------------
// 3) fp32 WMMA GEMM: C[M,N] = A[M,KDIM] * B[N,KDIM]^T   (K fixed = 200)
//    M % 16 == 0. N arbitrary.
//    256 threads = 8 waves; each wave owns TWO adjacent 16x16 C tiles
//    (two independent accumulator chains sharing one A fragment); the
//    16-row A tile is staged in LDS (float2 copies) and shared by all
//    8 waves.
//
//    Out-of-range N columns: addresses are CLAMPED (not masked) so the
//    loop is branch-free with EXEC all-ones; garbage columns are simply
//    never stored.
//
//    V_WMMA_F32_16X16X4_F32 layouts (ISA 7.12.2):
//      A 16x4: M = lane&15, K = 2*(lane>>4) + v   (v2f per lane)
//      B 4x16: N = lane&15, K = 2*(lane>>4) + v   (v2f per lane)
//      C/D:    vgpr r -> M = r + 8*(lane>>4), N = lane&15
// -------------------------------------------------------------------------
__global__ void __launch_bounds__(256)
wmma_gemm(const float* __restrict__ A, const float* __restrict__ Bm,
          float* __restrict__ C, int N) {
    __shared__ float As[16 * KDIM];

    const int m_tile = blockIdx.y * 16;
    const int wave   = threadIdx.x >> 5;
    const int lane   = threadIdx.x & 31;

    // Stage 16 x KDIM A tile as float2 (rows 8B-aligned: KDIM=200 even).
    {
        const v2f* A2 = (const v2f*)(A + (size_t)m_tile * KDIM);
        v2f*      As2 = (v2f*)As;
        constexpr int NV2 = 16 * KDIM / 2;                 // 1600
#pragma unroll 2
        for (int i = threadIdx.x; i < NV2; i += 256)
            As2[i] = A2[i];
    }
    __syncthreads();

    const int n_base = (blockIdx.x * 16 + wave * 2) * 16;  // 2 tiles / wave
    if (n_base >= N) return;                               // wave-uniform

    const int rowa  = lane & 15;
    const int khalf = (lane >> 4) * 2;
    const int n0    = n_base + (lane & 15);
    const int n1    = n0 + 16;
    const int n0c   = (n0 < N) ? n0 : (N - 1);             // clamp, don't mask
    const int n1c   = (n1 < N) ? n1 : (N - 1);
    const float* brow0 = Bm + (size_t)n0c * KDIM;
    const float* brow1 = Bm + (size_t)n1c * KDIM;
    __builtin_prefetch(brow0, 0, 3);
    __builtin_prefetch(brow1, 0, 3);

    const float* arow = &As[rowa * KDIM];

    v8f acc0 = {};
    v8f acc1 = {};
#pragma unroll 2
    for (int k0 = 0; k0 < KDIM; k0 += 4) {
        const int kk = k0 + khalf;                     // even -> 8B aligned
        const v2f a  = *(const v2f*)(arow + kk);       // ds_load b64
        const v2f b0 = *(const v2f*)(brow0 + kk);      // global_load_b64
        const v2f b1 = *(const v2f*)(brow1 + kk);
        acc0 = __builtin_amdgcn_wmma_f32_16x16x4_f32(
            false, a, false, b0, (short)0, acc0, false, false);
        acc1 = __builtin_amdgcn_wmma_f32_16x16x4_f32(
            false, a, false, b1, (short)0, acc1, false, false);
    }

    const int mrow = m_tile + (lane >> 4) * 8;
    if (n0 < N) {
#pragma unroll
        for (int r = 0; r < 8; ++r)
            C[(size_t)(mrow + r) * N + n0] = acc0[r];
    }
    if (n1 < N) {
#pragma unroll
        for (int r = 0; r < 8; ++r)
            C[(size_t)(mrow + r) * N + n1] = acc1[r];
    }
}

// -------------------------------------------------------------------------
// Launch
// -------------------------------------------------------------------------
extern "C" void kernel_launch(void* const* d_in, const int* in_sizes, int n_in,
                              void* d_out, int out_size, void* d_ws, size_t ws_size,
                              hipStream_t stream) {
    const int*   x      = (const int*)  d_in[0];
    const float* hidden = (const float*)d_in[1];
    const float* Wih0f  = (const float*)d_in[2];
    const float* Whh0f  = (const float*)d_in[3];
    const float* Wih0b  = (const float*)d_in[4];
    const float* Whh0b  = (const float*)d_in[5];
    const float* Wih1f  = (const float*)d_in[6];
    const float* Whh1f  = (const float*)d_in[7];
    const float* Wih1b  = (const float*)d_in[8];
    const float* Whh1b  = (const float*)d_in[9];
    const float* Wlin   = (const float*)d_in[10];
    float* out = (float*)d_out;

    // workspace layout (floats)
    float* ws   = (float*)d_ws;
    float* gx0f = ws;                    // BT*300
    float* gx0b = gx0f + BT * Gq;        // BT*300
    float* y0   = gx0b + BT * Gq;        // BT*200
    float* gx1f = y0   + BT * KDIM;      // BT*300
    float* gx1b = gx1f + BT * Gq;        // BT*300
    float* y1   = gx1b + BT * Gq;        // BT*200

    float* pred = out;                       // [BT][V]
    float* hT   = out + (size_t)BT * Vq;     // [4][B][H]

    // 1) layer-0 input projections (gather)
    const int n1 = BT * Gq;
    embed_kernel<<<(n1 + 255) / 256, 256, 0, stream>>>(x, Wih0f, Wih0b, gx0f, gx0b);

    // 2) layer-0 GRU scans (fwd + bwd concurrently)
    gru_kernel<<<2, 512, 0, stream>>>(gx0f, gx0b, Whh0f, Whh0b,
                                      hidden, y0, hT);

    // 3) layer-1 input GEMMs: gx1 = y0 [640,200] x Wih1^T [200,300]
    dim3 g3((Gq + 255) / 256, BT / 16);
    wmma_gemm<<<g3, 256, 0, stream>>>(y0, Wih1f, gx1f, Gq);
    wmma_gemm<<<g3, 256, 0, stream>>>(y0, Wih1b, gx1b, Gq);

    // 4) layer-1 GRU scans
    gru_kernel<<<2, 512, 0, stream>>>(gx1f, gx1b, Whh1f, Whh1b,
                                      hidden + 2 * Bq * Hq, y1, hT + 2 * Bq * Hq);

    // 5) output projection: pred = y1 [640,200] x Wlin^T [200,32000]
    dim3 g5(Vq / 256, BT / 16);
    wmma_gemm<<<g5, 256, 0, stream>>>(y1, Wlin, pred, Vq);
}